// Ins_31559419691549
// MI455X (gfx1250) — compile-verified
//
#include <hip/hip_runtime.h>
#include <hip/hip_bf16.h>

typedef __attribute__((ext_vector_type(2))) float v2f;
typedef __attribute__((ext_vector_type(8))) float v8f;

// ---------------------------------------------------------------------------
// Kernel 1: extract attention keys  keys[i] = A[i*NC + bag_label]
// ---------------------------------------------------------------------------
__global__ void extract_keys_kernel(const float* __restrict__ A,
                                    const int* __restrict__ bag_label,
                                    float* __restrict__ keys, int N, int NC) {
    int i = blockIdx.x * blockDim.x + threadIdx.x;
    if (i < N) keys[i] = A[i * NC + *bag_label];
}

// ---------------------------------------------------------------------------
// Kernel 2: exact dual rank (descending + ascending) with top_k tie-breaks.
// rank_desc(i) = #{j : k[j] > k[i]} + #{j<i : k[j]==k[i]}
// rank_asc(i)  = #{j : k[j] < k[i]} + #{j<i : k[j]==k[i]}
// Ranks are permutations -> scatter gives sorted top/bottom selections.
// Key vector (400KB) is LDS-tiled; it lives in L2 (192MB) after first read.
// ---------------------------------------------------------------------------
#define RANK_TILE 1024

__global__ void rank_scatter_kernel(const float* __restrict__ keys,
                                    int* __restrict__ idxbuf,
                                    int N, int n_ins) {
    __shared__ float tile[RANK_TILE];
    const int i = blockIdx.x * blockDim.x + threadIdx.x;
    const float my = (i < N) ? keys[i] : 0.0f;
    int gt = 0, lt = 0, eqb = 0;
    for (int t = 0; t < N; t += RANK_TILE) {
        const int cnt = min(RANK_TILE, N - t);
        for (int j = threadIdx.x; j < cnt; j += blockDim.x)
            tile[j] = keys[t + j];
        __syncthreads();
        if (i < N) {
#pragma unroll 8
            for (int j = 0; j < cnt; ++j) {
                const float v = tile[j];
                gt  += (v > my) ? 1 : 0;
                lt  += (v < my) ? 1 : 0;
                eqb += ((v == my) && ((t + j) < i)) ? 1 : 0;
            }
        }
        __syncthreads();
    }
    if (i < N) {
        const int dr = gt + eqb;            // position in descending sort
        if (dr < n_ins) idxbuf[dr] = i;     // pos_idx region
        const int ar = lt + eqb;            // position in ascending sort
        if (ar < n_ins) idxbuf[n_ins + ar] = i;  // neg_idx region
    }
}

// ---------------------------------------------------------------------------
// Kernel 3: instance labels (first n_ins -> 1, next n_ins -> 0)
// ---------------------------------------------------------------------------
__global__ void labels_kernel(float* __restrict__ out, int n_ins) {
    int r = blockIdx.x * blockDim.x + threadIdx.x;
    if (r < 2 * n_ins) out[r] = (r < n_ins) ? 1.0f : 0.0f;
}

// ---------------------------------------------------------------------------
// Kernel 4: gathered GEMM + bias + softmax via V_WMMA_F32_16X16X4_F32.
//   One wave32 computes a 16-row tile: D/4 chained fp32 WMMAs (K=4 each).
//   A 16x4 layout : lane l%16 = row M; lane half (l/16) selects K {0,1}/{2,3}
//   B 4x16 layout : lane l%16 = col N; lane half selects K rows (mirrors A)
//   C/D layout    : VGPR r -> row (r + 8*(lane/16)), col = lane%16
//   B is W zero-padded to 16 columns, staged in LDS (D*16 floats = 32KB).
// ---------------------------------------------------------------------------
__global__ void gemm_wmma_kernel(const float* __restrict__ h,
                                 const float* __restrict__ W,
                                 const float* __restrict__ bias,
                                 const int* __restrict__ idxbuf,
                                 float* __restrict__ out_unnorm,
                                 float* __restrict__ out_soft,
                                 int D, int NC, int total_rows) {
    extern __shared__ float Bp[];          // [D][16] zero-padded B operand
    const int tid = threadIdx.x;
    for (int e = tid; e < D * 16; e += blockDim.x) {
        const int k = e >> 4, n = e & 15;
        Bp[e] = (n < NC) ? W[k * NC + n] : 0.0f;
    }
    __syncthreads();

    const int wave = tid >> 5;             // 8 waves / block
    const int lane = tid & 31;
    const int l16  = lane & 15;
    const int half = lane >> 4;            // 0 or 1
    const int koff = half * 2;             // K sub-offset for this lane half
    const int rowbase = (blockIdx.x * (blockDim.x >> 5) + wave) * 16;

    const int myrow = rowbase + l16;
    const int src   = idxbuf[(myrow < total_rows) ? myrow : 0];
    const float* __restrict__ hrow = h + (long long)src * D;

    v8f acc = {0.f, 0.f, 0.f, 0.f, 0.f, 0.f, 0.f, 0.f};
    for (int k = 0; k < D; k += 4) {
        v2f a, b;
        a.x = hrow[k + koff];
        a.y = hrow[k + koff + 1];
        b.x = Bp[(k + koff) * 16 + l16];
        b.y = Bp[(k + koff + 1) * 16 + l16];
        // D = A(16x4 f32) * B(4x16 f32) + C  -> v_wmma_f32_16x16x4_f32
        acc = __builtin_amdgcn_wmma_f32_16x16x4_f32(
            /*neg_a=*/false, a, /*neg_b=*/false, b,
            /*c_mod=*/(short)0, acc, /*reuse_a=*/false, /*reuse_b=*/false);
    }

    const float bv = (l16 < NC) ? bias[l16] : 0.0f;
#pragma unroll
    for (int r = 0; r < 8; ++r) {
        const int row = rowbase + r + 8 * half;
        const float val = acc[r] + bv;
        // 2-class softmax: exchange with the adjacent column lane (wave32)
        const float other = __shfl_xor(val, 1, 32);
        const float m  = fmaxf(val, other);
        const float e0 = expf(val - m);
        const float e1 = expf(other - m);
        const float p  = e0 / (e0 + e1);
        if (row < total_rows && l16 < NC) {
            out_unnorm[row * NC + l16] = val;
            out_soft[row * NC + l16]   = p;
        }
    }
}

// ---------------------------------------------------------------------------
// Launcher
// ---------------------------------------------------------------------------
extern "C" void kernel_launch(void* const* d_in, const int* in_sizes, int n_in,
                              void* d_out, int out_size, void* d_ws, size_t ws_size,
                              hipStream_t stream) {
    const float* h   = (const float*)d_in[0];   // (N, D)
    const float* A   = (const float*)d_in[1];   // (N, 1, NC)
    const float* W   = (const float*)d_in[2];   // (D, NC)
    const float* b   = (const float*)d_in[3];   // (NC,)
    const int*   bag = (const int*)d_in[4];     // scalar

    const int NC    = in_sizes[3];
    const int D     = in_sizes[2] / NC;
    const int N     = in_sizes[0] / D;
    const int n_ins = N / 5;                    // TOPK_PCT = 0.2
    const int total = 2 * n_ins;

    // workspace: [keys: N floats][idxbuf: total ints]
    float* keys   = (float*)d_ws;
    int*   idxbuf = (int*)((char*)d_ws + (((size_t)N * sizeof(float) + 255) & ~(size_t)255));

    float* out_labels = (float*)d_out;                  // total
    float* out_unnorm = out_labels + total;             // total*NC
    float* out_soft   = out_unnorm + (size_t)total * NC;// total*NC

    const int threads = 256;

    extract_keys_kernel<<<(N + threads - 1) / threads, threads, 0, stream>>>(
        A, bag, keys, N, NC);

    rank_scatter_kernel<<<(N + threads - 1) / threads, threads, 0, stream>>>(
        keys, idxbuf, N, n_ins);

    labels_kernel<<<(total + threads - 1) / threads, threads, 0, stream>>>(
        out_labels, n_ins);

    const int rows_per_block = (threads / 32) * 16;     // 128
    const int gblocks = (total + rows_per_block - 1) / rows_per_block;
    const size_t shmem = (size_t)D * 16 * sizeof(float);
    gemm_wmma_kernel<<<gblocks, threads, shmem, stream>>>(
        h, W, b, idxbuf, out_unnorm, out_soft, D, NC, total);
}